// PeripheralEncoder_4423816315040
// MI455X (gfx1250) — compile-verified
//
#include <hip/hip_runtime.h>

typedef __attribute__((ext_vector_type(2))) float v2f;
typedef __attribute__((ext_vector_type(8))) float v8f;

#define D      1024
#define BS     32
#define NB     256
#define M_ROWS 1024   /* B*NB = 4*256 */
#define K3     3072   /* 3*D */

// ---------------------------------------------------------------------------
// Kernel 1: per-block softmax-weighted mean/std + max  ->  feat [M_ROWS, 3072]
// One workgroup per (batch, block): 32 rows x 1024 cols = 128 KB read.
// ---------------------------------------------------------------------------
__global__ __launch_bounds__(256) void k1_blockstats(
    const float* __restrict__ x, const float* __restrict__ w_w,
    const float* __restrict__ w_b, float* __restrict__ feat)
{
    __shared__ float logits_s[BS];
    __shared__ float wt_s[BS];

    const int bi   = blockIdx.x;        // 0..1023 == b*NB + blk
    const int t    = threadIdx.x;
    const int lane = t & 31;
    const int wave = t >> 5;

    const float4* __restrict__ ww4 = (const float4*)w_w;

    // ---- Phase 1: logits[r] = dot(x[r,:], w_w) + w_b. 8 waves x 4 rows. ----
    #pragma unroll
    for (int i = 0; i < 4; ++i) {
        const int r = wave * 4 + i;
        const float4* xr4 = (const float4*)(x + (size_t)(bi * BS + r) * D);
        float acc = 0.f;
        #pragma unroll
        for (int j = 0; j < 8; ++j) {
            float4 xv = xr4[lane + 32 * j];
            float4 wv = ww4[lane + 32 * j];
            acc += xv.x * wv.x + xv.y * wv.y + xv.z * wv.z + xv.w * wv.w;
        }
        #pragma unroll
        for (int off = 16; off > 0; off >>= 1)
            acc += __shfl_xor(acc, off, 32);
        if (lane == 0) logits_s[r] = acc + w_b[0];
    }
    __syncthreads();

    // ---- Phase 2: softmax over the 32 logits (wave 0 only) ----
    if (t < 32) {
        float l = logits_s[t];
        float m = l;
        #pragma unroll
        for (int off = 16; off > 0; off >>= 1)
            m = fmaxf(m, __shfl_xor(m, off, 32));
        float e = expf(l - m);
        float s = e;
        #pragma unroll
        for (int off = 16; off > 0; off >>= 1)
            s += __shfl_xor(s, off, 32);
        wt_s[t] = e / s;
    }
    __syncthreads();

    // ---- Phase 3: per-column weighted stats; 4 cols (one float4) / thread ----
    const float4* xb4 = (const float4*)(x + (size_t)bi * BS * D);
    float4 mn = {0.f, 0.f, 0.f, 0.f};
    float4 s2 = {0.f, 0.f, 0.f, 0.f};
    float4 mx = {-3.402823466e38f, -3.402823466e38f,
                 -3.402823466e38f, -3.402823466e38f};
    #pragma unroll 4
    for (int k = 0; k < BS; ++k) {
        const float w = wt_s[k];
        float4 v = xb4[k * (D / 4) + t];
        mn.x += w * v.x;  mn.y += w * v.y;  mn.z += w * v.z;  mn.w += w * v.w;
        s2.x += w * v.x * v.x;  s2.y += w * v.y * v.y;
        s2.z += w * v.z * v.z;  s2.w += w * v.w * v.w;
        mx.x = fmaxf(mx.x, v.x);  mx.y = fmaxf(mx.y, v.y);
        mx.z = fmaxf(mx.z, v.z);  mx.w = fmaxf(mx.w, v.w);
    }
    float4 sd;
    sd.x = sqrtf(fmaxf(s2.x - mn.x * mn.x, 0.f) + 1e-8f);
    sd.y = sqrtf(fmaxf(s2.y - mn.y * mn.y, 0.f) + 1e-8f);
    sd.z = sqrtf(fmaxf(s2.z - mn.z * mn.z, 0.f) + 1e-8f);
    sd.w = sqrtf(fmaxf(s2.w - mn.w * mn.w, 0.f) + 1e-8f);

    float4* f4 = (float4*)(feat + (size_t)bi * K3);
    f4[t]            = mn;   // feat[:,    0:1024] = mean
    f4[256 + t]      = sd;   // feat[:, 1024:2048] = std
    f4[512 + t]      = mx;   // feat[:, 2048:3072] = max
}

// ---------------------------------------------------------------------------
// Kernel 2: h = feat @ proj_w + proj_b  via V_WMMA_F32_16X16X4_F32
// 128 threads (4 waves), 64x64 tile per WG, waves in 2x2 of 32x32 (2x2 WMMA).
// ---------------------------------------------------------------------------
__global__ __launch_bounds__(128) void k2_gemm(
    const float* __restrict__ feat, const float* __restrict__ pw,
    const float* __restrict__ pb, float* __restrict__ h)
{
    // Pad strides: As stride 36 -> A-frag column reads conflict-free;
    // Bs stride 72 -> rows kk and kk+2 land 16 banks apart (half-wave split).
    __shared__ float As[64][36];
    __shared__ float Bs[32][72];

    const int t     = threadIdx.x;
    const int lane  = t & 31;
    const int wave  = t >> 5;
    const int tileM = blockIdx.x * 64;
    const int tileN = blockIdx.y * 64;
    const int waveM = (wave >> 1) * 32;
    const int waveN = (wave & 1)  * 32;
    const int llo   = lane & 15;        // row/col within 16
    const int lhi   = lane >> 4;        // half-wave select

    v8f acc[2][2] = {};

    for (int k0 = 0; k0 < K3; k0 += 32) {
        __syncthreads();
        // Stage As[64][32] = feat[tileM.., k0..]; 512 float4, 4/thread.
        #pragma unroll
        for (int i = 0; i < 4; ++i) {
            int f = t + 128 * i;
            int r = f >> 3, cg = f & 7;
            float4 v = *(const float4*)(feat + (size_t)(tileM + r) * K3 + k0 + cg * 4);
            *(float4*)&As[r][cg * 4] = v;
        }
        // Stage Bs[32][64] = proj_w[k0.., tileN..]; 512 float4, 4/thread.
        #pragma unroll
        for (int i = 0; i < 4; ++i) {
            int f = t + 128 * i;
            int r = f >> 4, cg = f & 15;
            float4 v = *(const float4*)(pw + (size_t)(k0 + r) * D + tileN + cg * 4);
            *(float4*)&Bs[r][cg * 4] = v;
        }
        __syncthreads();

        #pragma unroll
        for (int kk = 0; kk < 32; kk += 4) {
            const int kb = kk + (lhi << 1);   // lanes 0-15: K={kk,kk+1}; 16-31: K={kk+2,kk+3}
            v2f a0 = { As[waveM + llo][kb],      As[waveM + llo][kb + 1] };
            v2f a1 = { As[waveM + 16 + llo][kb], As[waveM + 16 + llo][kb + 1] };
            v2f b0 = { Bs[kb][waveN + llo],      Bs[kb + 1][waveN + llo] };
            v2f b1 = { Bs[kb][waveN + 16 + llo], Bs[kb + 1][waveN + 16 + llo] };
            acc[0][0] = __builtin_amdgcn_wmma_f32_16x16x4_f32(
                false, a0, false, b0, (short)0, acc[0][0], false, false);
            acc[0][1] = __builtin_amdgcn_wmma_f32_16x16x4_f32(
                false, a0, false, b1, (short)0, acc[0][1], false, false);
            acc[1][0] = __builtin_amdgcn_wmma_f32_16x16x4_f32(
                false, a1, false, b0, (short)0, acc[1][0], false, false);
            acc[1][1] = __builtin_amdgcn_wmma_f32_16x16x4_f32(
                false, a1, false, b1, (short)0, acc[1][1], false, false);
        }
    }

    // Epilogue: D layout: VGPR i, lanes 0-15 -> M=i, N=lane; lanes 16-31 -> M=i+8.
    #pragma unroll
    for (int mt = 0; mt < 2; ++mt) {
        #pragma unroll
        for (int nt = 0; nt < 2; ++nt) {
            const int col = tileN + waveN + nt * 16 + llo;
            const float pbv = pb[col];
            #pragma unroll
            for (int i = 0; i < 8; ++i) {
                const int row = tileM + waveM + mt * 16 + (lhi << 3) + i;
                h[(size_t)row * D + col] = acc[mt][nt][i] + pbv;
            }
        }
    }
}

// ---------------------------------------------------------------------------
// Kernel 3: LayerNorm over D + pos add. One WG per row.
// ---------------------------------------------------------------------------
__global__ __launch_bounds__(256) void k3_ln(
    const float* __restrict__ h, const float* __restrict__ g,
    const float* __restrict__ b, const float* __restrict__ pos,
    float* __restrict__ out)
{
    __shared__ float red_s[8], red2_s[8];
    __shared__ float mu_s, rs_s;

    const int row  = blockIdx.x;
    const int t    = threadIdx.x;
    const int lane = t & 31;
    const int wave = t >> 5;

    float4 v = ((const float4*)(h + (size_t)row * D))[t];
    float s  = v.x + v.y + v.z + v.w;
    float s2 = v.x * v.x + v.y * v.y + v.z * v.z + v.w * v.w;
    #pragma unroll
    for (int off = 16; off > 0; off >>= 1) {
        s  += __shfl_xor(s, off, 32);
        s2 += __shfl_xor(s2, off, 32);
    }
    if (lane == 0) { red_s[wave] = s; red2_s[wave] = s2; }
    __syncthreads();
    if (t == 0) {
        float ts = 0.f, ts2 = 0.f;
        #pragma unroll
        for (int i = 0; i < 8; ++i) { ts += red_s[i]; ts2 += red2_s[i]; }
        const float mu  = ts * (1.0f / D);
        const float var = ts2 * (1.0f / D) - mu * mu;
        mu_s = mu;
        rs_s = 1.0f / sqrtf(var + 1e-5f);
    }
    __syncthreads();
    const float mu = mu_s, rs = rs_s;

    float4 gv = ((const float4*)g)[t];
    float4 bv = ((const float4*)b)[t];
    const int blk = row & (NB - 1);
    float4 pv = ((const float4*)(pos + (size_t)blk * D))[t];

    float4 o;
    o.x = (v.x - mu) * rs * gv.x + bv.x + pv.x;
    o.y = (v.y - mu) * rs * gv.y + bv.y + pv.y;
    o.z = (v.z - mu) * rs * gv.z + bv.z + pv.z;
    o.w = (v.w - mu) * rs * gv.w + bv.w + pv.w;
    ((float4*)(out + (size_t)row * D))[t] = o;
}

// ---------------------------------------------------------------------------
extern "C" void kernel_launch(void* const* d_in, const int* in_sizes, int n_in,
                              void* d_out, int out_size, void* d_ws, size_t ws_size,
                              hipStream_t stream) {
    const float* x      = (const float*)d_in[0];
    const float* w_w    = (const float*)d_in[1];
    const float* w_b    = (const float*)d_in[2];
    const float* proj_w = (const float*)d_in[3];
    const float* proj_b = (const float*)d_in[4];
    const float* ln_g   = (const float*)d_in[5];
    const float* ln_b   = (const float*)d_in[6];
    const float* pos    = (const float*)d_in[7];
    float* out  = (float*)d_out;

    float* feat = (float*)d_ws;                       // M_ROWS * K3 floats (12 MB)
    float* hbuf = feat + (size_t)M_ROWS * K3;         // M_ROWS * D  floats (4 MB)

    k1_blockstats<<<M_ROWS, 256, 0, stream>>>(x, w_w, w_b, feat);
    k2_gemm<<<dim3(M_ROWS / 64, D / 64), 128, 0, stream>>>(feat, proj_w, proj_b, hbuf);
    k3_ln<<<M_ROWS, 256, 0, stream>>>(hbuf, ln_g, ln_b, pos, out);
}